// Word2VecMean_75617194213687
// MI455X (gfx1250) — compile-verified
//
#include <hip/hip_runtime.h>

// CDNA5 / gfx1250: wave32, WMMA f32 16x16x4 used as an f32-exact
// indicator-matrix segment reducer.  D = A(16x4 membership 0/1) x B(4x16 emb).

typedef __attribute__((ext_vector_type(2))) float v2f;
typedef __attribute__((ext_vector_type(8))) float v8f;

namespace {
constexpr int kEmb          = 128;  // embedding width (reference EMBED)
constexpr int kSegsPerGroup = 16;   // one WMMA tile of segments
constexpr int kColSplit     = 4;    // 4 waves per group -> 32 cols each
constexpr int kWavesPerBlk  = 8;    // 256 threads / wave32
}

__global__ __launch_bounds__(256)
void word2vec_mean_wmma(const int* __restrict__ tok,
                        const int* __restrict__ seg,
                        const float* __restrict__ emb,
                        float* __restrict__ out,
                        int T, int nSeg, int nGroups)
{
    const int lane   = threadIdx.x & 31;
    const int wave   = threadIdx.x >> 5;
    const int waveId = blockIdx.x * kWavesPerBlk + wave;

    int group         = waveId >> 2;              // / kColSplit
    const int cchunk  = waveId & (kColSplit - 1); // which 32-column slice
    const bool valid  = (group < nGroups);
    if (!valid) group = nGroups - 1;              // keep all waves in barriers
    const int segBase = group * kSegsPerGroup;

    // ---- per-wave segment bounds via binary search (segment_ids is sorted) ----
    __shared__ int sBounds[kWavesPerBlk][kSegsPerGroup + 1];
    if (lane <= kSegsPerGroup) {
        const int v = segBase + lane;             // lower_bound(seg, v)
        int lo = 0, hi = T;
        while (lo < hi) {
            const int mid = (lo + hi) >> 1;
            if (seg[mid] < v) lo = mid + 1; else hi = mid;
        }
        sBounds[wave][lane] = lo;
    }
    __syncthreads();
    // Wave-uniform loop bounds -> SALU loop control.
    const int start = __builtin_amdgcn_readfirstlane(sBounds[wave][0]);
    const int end   = __builtin_amdgcn_readfirstlane(sBounds[wave][kSegsPerGroup]);

    // ---- WMMA lane coordinates ----
    // A 16x4 f32: VGPR0 -> K = (lane<16 ? 0 : 2), VGPR1 -> K+1, M = lane&15.
    // B 4x16 f32: VGPR0 -> row K = (lane<16 ? 0 : 2), VGPR1 -> K+1, N = lane&15.
    const int m         = lane & 15;
    const int kx        = (lane < 16) ? 0 : 2;    // K index for .x regs
    const int ky        = kx + 1;                 // K index for .y regs
    const int colBase   = cchunk * 32;            // two 16-col blocks
    const int segTarget = segBase + m;
    const int Tm1       = T - 1;
    const unsigned colOff = (unsigned)(colBase + m);

    v8f acc0 = {};                                // cols [colBase, colBase+16)
    v8f acc1 = {};                                // cols [colBase+16, colBase+32)

    #pragma unroll 4
    for (int t = start; t < end; t += 4) {
        const int ix  = t + kx;
        const int iy  = t + ky;
        // Branchless tail handling: always load (clamped), mask in compare.
        const int ixc = (ix < Tm1) ? ix : Tm1;
        const int iyc = (iy < Tm1) ? iy : Tm1;

        const int sx  = seg[ixc];
        const int sy  = seg[iyc];
        const int tkx = tok[ixc];
        const int tky = tok[iyc];

        v2f a;
        a.x = (ix < end && sx == segTarget) ? 1.0f : 0.0f;
        a.y = (iy < end && sy == segTarget) ? 1.0f : 0.0f;

        // 32-bit element offsets (table is 12.8M floats < 2^31): saddr form.
        const unsigned ox = (unsigned)tkx * 128u + colOff;
        const unsigned oy = (unsigned)tky * 128u + colOff;

        v2f b0, b1;
        b0.x = emb[ox];
        b0.y = emb[oy];
        b1.x = emb[ox + 16u];
        b1.y = emb[oy + 16u];

        // D = A x B + C   (f32, round-to-nearest-even, exact vs reference path)
        acc0 = __builtin_amdgcn_wmma_f32_16x16x4_f32(
                   false, a, false, b0, (short)0, acc0, false, false);
        acc1 = __builtin_amdgcn_wmma_f32_16x16x4_f32(
                   false, a, false, b1, (short)0, acc1, false, false);
    }

    // ---- divide by counts and store ----
    // C/D layout: VGPR r holds M = r + (lane<16 ? 0 : 8), N = lane&15.
    const int half = (lane < 16) ? 0 : 8;
    if (valid) {
        #pragma unroll
        for (int r = 0; r < 8; ++r) {
            const int mm = r + half;
            const int s  = segBase + mm;
            if (s < nSeg) {
                const int cnt   = sBounds[wave][mm + 1] - sBounds[wave][mm];
                const float inv = 1.0f / (float)(cnt > 1 ? cnt : 1);
                const size_t o  = (size_t)s * kEmb + colBase + m;
                out[o]      = acc0[r] * inv;
                out[o + 16] = acc1[r] * inv;
            }
        }
    }
}

extern "C" void kernel_launch(void* const* d_in, const int* in_sizes, int n_in,
                              void* d_out, int out_size, void* d_ws, size_t ws_size,
                              hipStream_t stream)
{
    (void)n_in; (void)d_ws; (void)ws_size;
    const int*   tok = (const int*)d_in[0];   // token_ids  [T] int32
    const int*   seg = (const int*)d_in[1];   // segment_ids[T] int32 (sorted)
    const float* emb = (const float*)d_in[2]; // embedding  [VOCAB,128] f32
    float*       out = (float*)d_out;         // means      [B,128] f32

    const int T       = in_sizes[0];
    const int nSeg    = out_size / kEmb;                  // 16384
    const int nGroups = (nSeg + kSegsPerGroup - 1) / kSegsPerGroup;
    const int waves   = nGroups * kColSplit;              // 4096
    const int blocks  = (waves + kWavesPerBlk - 1) / kWavesPerBlk;

    word2vec_mean_wmma<<<blocks, kWavesPerBlk * 32, 0, stream>>>(
        tok, seg, emb, out, T, nSeg, nGroups);
}